// FunctionalFlow_25907242729814
// MI455X (gfx1250) — compile-verified
//
#include <hip/hip_runtime.h>

// CDNA5 / gfx1250, wave32. Exact-fp32 WMMA reduction over in_channels,
// K split across wave pairs, branchless tanh (v_exp_f32 + v_rcp_f32).

typedef float v2f __attribute__((ext_vector_type(2)));
typedef float v8f __attribute__((ext_vector_type(8)));

#define IN_CH 64
#define OUT_CH 256
#define NUM_POINTS 16
#define TILE_ROWS 16
#define WAVES_PER_BLOCK 8
#define TILES_PER_BLOCK (WAVES_PER_BLOCK / 2)   // 2 waves cooperate per tile
#define BLOCK_THREADS (WAVES_PER_BLOCK * 32)

__device__ __forceinline__ float fast_exp2(float y) {
#if __has_builtin(__builtin_amdgcn_exp2f)
  return __builtin_amdgcn_exp2f(y);             // v_exp_f32
#else
  return exp2f(y);
#endif
}

__device__ __forceinline__ float fast_rcp(float y) {
#if __has_builtin(__builtin_amdgcn_rcpf)
  return __builtin_amdgcn_rcpf(y);              // v_rcp_f32
#else
  return 1.0f / y;
#endif
}

// Branchless tanh: t = 1 - 2/(exp(2x)+1). Exact limits at +/-inf, no NaN.
__device__ __forceinline__ float fast_tanh(float x) {
  float e = fast_exp2(x * 2.8853900817779268f); // exp(2x) = 2^(2*log2(e)*x)
  float r = fast_rcp(e + 1.0f);
  return __builtin_fmaf(-2.0f, r, 1.0f);
}

// 3-step tanh flow on a scalar. lut[p] = (velo*cos(angles), velo*sin(angles)).
__device__ __forceinline__ float flow3(float x, const float2* lut) {
#pragma unroll
  for (int s = 0; s < 3; ++s) {
    float t = fast_tanh(x);
    int pos = (int)rintf(__builtin_fmaf(t, 8.0f, 8.0f));   // RNE == jnp.round
    pos = pos < 0 ? 0 : (pos > NUM_POINTS - 1 ? NUM_POINTS - 1 : pos);
    float2 e = lut[pos];                                   // ds_load_b64 gather
    // x += (v*cos(a) + t*v*sin(a)) / NUM_STEPS
    x = __builtin_fmaf(__builtin_fmaf(t, e.y, e.x), (1.0f / 3.0f), x);
  }
  return x;
}

__global__ void __launch_bounds__(BLOCK_THREADS, 2)
flow_wmma_kernel(const float* __restrict__ data,
                 const float* __restrict__ angles,
                 const float* __restrict__ velo,
                 float* __restrict__ out,
                 int nTiles) {
  __shared__ float2 lut[NUM_POINTS];
  __shared__ float part[WAVES_PER_BLOCK][8][32];  // partial D tiles, 8 KB

  const int tid = threadIdx.x;
  if (tid < NUM_POINTS) {
    float a = angles[tid];
    float v = velo[tid];
    float sa, ca;
    sincosf(a, &sa, &ca);
    lut[tid] = make_float2(v * ca, v * sa);
  }
  __syncthreads();

  const int lane = tid & 31;
  const int wave = tid >> 5;
  const int pairId = wave >> 1;                  // tile within block
  const int khalf = wave & 1;                    // which 32-channel half
  const int tile = blockIdx.x * TILES_PER_BLOCK + pairId;
  const bool active = tile < nTiles;             // wave-uniform -> scalar branch
  const int n0 = tile * TILE_ROWS;

  v8f acc = {};                                  // C/D 16x16 f32, 8 VGPRs
  const v2f bones = {1.0f, 1.0f};                // B = ones(4x16)

  if (active) {
    // fp32 A-matrix (16x4) layout: lanes 0-15 hold K={0,1} of row M=lane,
    // lanes 16-31 hold K={2,3} of row M=lane-16 -> per-lane float2 load.
    const int row = n0 + (lane & 15);
    const int koff = (lane >> 4) << 1;           // 0 or 2
    const float* rp = data + (size_t)row * IN_CH + khalf * (IN_CH / 2) + koff;
#pragma unroll
    for (int kc = 0; kc < IN_CH / 2; kc += 4) {  // 8 chunks of K=4
      float2 a2 = *(const float2*)(rp + kc);     // global_load_b64
      v2f av;
      av.x = flow3(a2.x, lut);
      av.y = flow3(a2.y, lut);
      // D = A(16x4) x ones(4x16) + C => per-row partial channel sums
      acc = __builtin_amdgcn_wmma_f32_16x16x4_f32(
          /*neg_a=*/false, av, /*neg_b=*/false, bones,
          /*c_mod=*/(short)0, acc, /*reuse_a=*/false, /*reuse_b=*/false);
    }
  }

  // Exchange partial accumulators between the wave pair via LDS.
#pragma unroll
  for (int r = 0; r < 8; ++r) part[wave][r][lane] = acc[r];  // conflict-free
  __syncthreads();
  const int partner = wave ^ 1;
#pragma unroll
  for (int r = 0; r < 8; ++r) acc[r] += part[partner][r][lane];

  if (active) {
    // D layout: VGPR r, lanes 0-15 -> row n0+r; lanes 16-31 -> row n0+r+8.
    // Every column equals the row sum; each wave of the pair broadcasts it
    // across its half of the 256 output columns.
    const int rbase = (lane >> 4) << 3;          // 0 or 8
    const int col = khalf * (OUT_CH / 2) + ((lane & 15) << 2);
#pragma unroll
    for (int r = 0; r < 8; ++r) {
      float v = acc[r];
      float4 v4 = make_float4(v, v, v, v);
      float* p = out + (size_t)(n0 + rbase + r) * OUT_CH + col;
#pragma unroll
      for (int rep = 0; rep < 2; ++rep) {
        *reinterpret_cast<float4*>(p + rep * 64) = v4;  // global_store_b128
      }
    }
  }
}

extern "C" void kernel_launch(void* const* d_in, const int* in_sizes, int n_in,
                              void* d_out, int out_size, void* d_ws, size_t ws_size,
                              hipStream_t stream) {
  const float* data   = (const float*)d_in[0];
  const float* angles = (const float*)d_in[1];
  const float* velo   = (const float*)d_in[2];
  float* out = (float*)d_out;

  const int N = in_sizes[0] / IN_CH;             // 4096
  const int nTiles = N / TILE_ROWS;              // 256 tiles of 16 rows
  const int blocks = (nTiles + TILES_PER_BLOCK - 1) / TILES_PER_BLOCK;  // 64

  flow_wmma_kernel<<<blocks, BLOCK_THREADS, 0, stream>>>(data, angles, velo, out, nTiles);
}